// Temporalselect_67688684584965
// MI455X (gfx1250) — compile-verified
//
#include <hip/hip_runtime.h>
#include <hip/hip_bf16.h>
#include <math.h>

typedef float v2f __attribute__((ext_vector_type(2)));
typedef float v8f __attribute__((ext_vector_type(8)));

#define BDIM   64
#define NDIM   1568
#define DDIM   768
#define GROUPS 8
#define SEG    196          // N / GROUPS
#define ROWS   512          // B * GROUPS
#define HALF_D 384
#define OUTR   393          // 1 + NUM_FRAMES*T/8 = 1 + 392
#define EPSV   768.0f

// ---------------------------------------------------------------------------
// Kernel 1: per-(b,g) max over 196 rows, then LayerNorm over D=768.
// 512 blocks x 192 threads, float4 (b128) streaming loads: the 308 MB read of
// x is the dominant cost of the whole problem.
// ---------------------------------------------------------------------------
__global__ void segmax_ln_kernel(const float* __restrict__ x,
                                 const float* __restrict__ ln_w,
                                 const float* __restrict__ ln_b,
                                 float* __restrict__ xn) {
  const int i = blockIdx.x;        // 0..511  (b*8+g)
  const int t = threadIdx.x;       // 0..191
  const float4* xp = (const float4*)(x + (size_t)i * SEG * DDIM);
  float4 m = make_float4(-3.402823e38f, -3.402823e38f, -3.402823e38f, -3.402823e38f);
  for (int r = 0; r < SEG; ++r) {
    float4 v = xp[r * (DDIM / 4) + t];
    m.x = fmaxf(m.x, v.x); m.y = fmaxf(m.y, v.y);
    m.z = fmaxf(m.z, v.z); m.w = fmaxf(m.w, v.w);
  }
  __shared__ float red[192];
  red[t] = m.x + m.y + m.z + m.w;
  __syncthreads();
  for (int off = 128; off > 0; off >>= 1) {
    if (t < off && t + off < 192) red[t] += red[t + off];
    __syncthreads();
  }
  const float mu = red[0] * (1.0f / 768.0f);
  __syncthreads();
  float4 d = make_float4(m.x - mu, m.y - mu, m.z - mu, m.w - mu);
  red[t] = d.x * d.x + d.y * d.y + d.z * d.z + d.w * d.w;
  __syncthreads();
  for (int off = 128; off > 0; off >>= 1) {
    if (t < off && t + off < 192) red[t] += red[t + off];
    __syncthreads();
  }
  const float var = red[0] * (1.0f / 768.0f);
  const float inv = 1.0f / sqrtf(var + EPSV);
  float4 w  = ((const float4*)ln_w)[t];
  float4 bb = ((const float4*)ln_b)[t];
  float4 o;
  o.x = d.x * inv * w.x + bb.x;
  o.y = d.y * inv * w.y + bb.y;
  o.z = d.z * inv * w.z + bb.z;
  o.w = d.w * inv * w.w + bb.w;
  ((float4*)(xn + (size_t)i * DDIM))[t] = o;
}

// ---------------------------------------------------------------------------
// Kernel 2: f32 WMMA GEMM  C[M,N] = A[M,K] @ B[K,N]  using V_WMMA_F32_16X16X4.
// One 16x16 C tile per wave; K stepped by 4. Per ISA 7.12.2 layouts:
//   A: reg v, lane l -> A[M = l&15,          K = k + 2*(l>>4) + v]
//   B: reg v, lane l -> B[K = k + 2*(l>>4)+v, N = l&15]
//   C: reg v, lane l -> C[M = v + 8*(l>>4),   N = l&15]
// ---------------------------------------------------------------------------
__global__ void wmma_gemm_f32(const float* __restrict__ A,
                              const float* __restrict__ Bm,
                              float* __restrict__ C,
                              int M, int N, int K) {
  const int gw   = (int)((blockIdx.x * blockDim.x + threadIdx.x) >> 5);
  const int lane = threadIdx.x & 31;
  const int tilesM = M >> 4;
  const int tM = (gw % tilesM) << 4;
  const int tN = (gw / tilesM) << 4;
  if (tN >= N) return;                 // wave-uniform: EXEC stays all-ones
  const int lo = lane & 15;
  const int hi = lane >> 4;

  const float* aRow = A + (size_t)(tM + lo) * K;
  const float* bCol = Bm + (tN + lo);

  v8f c = {};
  #pragma unroll 4
  for (int k = 0; k < K; k += 4) {
    const int kk = k + 2 * hi;
    const float2 av = *(const float2*)(aRow + kk);   // 8B aligned (kk even)
    v2f a; a.x = av.x; a.y = av.y;
    v2f b; b.x = bCol[(size_t)kk * N];
           b.y = bCol[(size_t)(kk + 1) * N];
    c = __builtin_amdgcn_wmma_f32_16x16x4_f32(
        /*neg_a=*/false, a, /*neg_b=*/false, b,
        /*c_mod=*/(short)0, c, /*reuse_a=*/false, /*reuse_b=*/false);
  }
  float* cBase = C + (size_t)(tM + 8 * hi) * N + tN + lo;
  #pragma unroll
  for (int v = 0; v < 8; ++v) cBase[(size_t)v * N] = c[v];
}

// ---------------------------------------------------------------------------
// Kernel 3: xc[i, 0:384] = x_m[i, :], xc[i, 384:768] = mean_g x_m[b(i), g, :]
// ---------------------------------------------------------------------------
__global__ void build_xc_kernel(const float* __restrict__ x_m,
                                float* __restrict__ xc) {
  const int idx = blockIdx.x * blockDim.x + threadIdx.x;   // 512*768 exact
  const int i = idx / DDIM;
  const int j = idx - i * DDIM;
  float v;
  if (j < HALF_D) {
    v = x_m[(size_t)i * HALF_D + j];
  } else {
    const int b  = i >> 3;
    const int jj = j - HALF_D;
    float sum = 0.0f;
    #pragma unroll
    for (int g = 0; g < GROUPS; ++g) sum += x_m[(size_t)(b * GROUPS + g) * HALF_D + jj];
    v = sum * 0.125f;
  }
  xc[idx] = v;
}

// ---------------------------------------------------------------------------
// Kernel 4: s[i] = t2[i,:] . w4
// ---------------------------------------------------------------------------
__global__ void dot_w4_kernel(const float* __restrict__ t2,
                              const float* __restrict__ w4,
                              float* __restrict__ s) {
  const int i = blockIdx.x;       // 0..511
  const int t = threadIdx.x;      // 0..255
  const float* row = t2 + (size_t)i * DDIM;
  float acc = row[t] * w4[t] + row[t + 256] * w4[t + 256] + row[t + 512] * w4[t + 512];
  __shared__ float red[256];
  red[t] = acc;
  __syncthreads();
  for (int off = 128; off > 0; off >>= 1) {
    if (t < off) red[t] += red[t + off];
    __syncthreads();
  }
  if (t == 0) s[i] = red[0];
}

// ---------------------------------------------------------------------------
// Kernel 5: top-2 indices per batch (min-max norm is monotone -> same order;
// first-occurrence argmax matches lax.top_k tie-breaking).
// ---------------------------------------------------------------------------
__global__ void top2_kernel(const float* __restrict__ s, int* __restrict__ idx) {
  const int b = blockIdx.x * blockDim.x + threadIdx.x;
  if (b >= BDIM) return;
  float best = -3.402823e38f; int bi = 0;
  #pragma unroll
  for (int g = 0; g < GROUPS; ++g) {
    float v = s[b * GROUPS + g];
    if (v > best) { best = v; bi = g; }
  }
  float best2 = -3.402823e38f; int bi2 = 0;
  #pragma unroll
  for (int g = 0; g < GROUPS; ++g) {
    if (g == bi) continue;
    float v = s[b * GROUPS + g];
    if (v > best2) { best2 = v; bi2 = g; }
  }
  idx[b * 2 + 0] = bi;
  idx[b * 2 + 1] = bi2;
}

// ---------------------------------------------------------------------------
// Kernel 6: output assembly. out[b,0,:] = inp[b,0,:];
// out[b, 1 + k*196 + r, :] = inp[b, 1 + idx[b,k]*196 + r, :]. float4 copies.
// ---------------------------------------------------------------------------
__global__ void gather_kernel(const float* __restrict__ inp,
                              const int* __restrict__ idx,
                              float* __restrict__ out) {
  const int row = blockIdx.x;            // 0 .. 64*393-1
  const int b = row / OUTR;
  const int r = row - b * OUTR;
  const int t = threadIdx.x;             // 0..191 (float4)
  int srcRow;
  if (r == 0) {
    srcRow = 0;
  } else {
    const int rp = r - 1;
    const int k  = rp / SEG;
    const int rr = rp - k * SEG;
    const int g  = idx[b * 2 + k];
    srcRow = 1 + g * SEG + rr;
  }
  const float4* src = (const float4*)(inp + (size_t)b * (NDIM + 1) * DDIM + (size_t)srcRow * DDIM);
  float4* dst = (float4*)(out + ((size_t)b * OUTR + r) * DDIM);
  dst[t] = src[t];
}

// ---------------------------------------------------------------------------
extern "C" void kernel_launch(void* const* d_in, const int* in_sizes, int n_in,
                              void* d_out, int out_size, void* d_ws, size_t ws_size,
                              hipStream_t stream) {
  const float* x    = (const float*)d_in[0];
  const float* inp  = (const float*)d_in[1];
  const float* ln_w = (const float*)d_in[2];
  const float* ln_b = (const float*)d_in[3];
  const float* w1   = (const float*)d_in[4];
  const float* w2   = (const float*)d_in[5];
  const float* w3   = (const float*)d_in[6];
  const float* w4   = (const float*)d_in[7];
  float* out = (float*)d_out;

  float* ws = (float*)d_ws;
  float* xn  = ws;                         // 512*768
  float* t1  = xn  + (size_t)ROWS * DDIM;  // 512*768
  float* xm  = t1  + (size_t)ROWS * DDIM;  // 512*384
  float* xc  = xm  + (size_t)ROWS * HALF_D;// 512*768
  float* t2  = xc  + (size_t)ROWS * DDIM;  // 512*768
  float* s   = t2  + (size_t)ROWS * DDIM;  // 512
  int*   idx = (int*)(s + ROWS);           // 128

  // 1) segment max + layernorm
  segmax_ln_kernel<<<ROWS, 192, 0, stream>>>(x, ln_w, ln_b, xn);

  // 2) t1 = xn @ w1   (512x768 @ 768x768): 32*48 = 1536 tiles, 8 waves/block
  wmma_gemm_f32<<<192, 256, 0, stream>>>(xn, w1, t1, ROWS, DDIM, DDIM);

  // 3) x_m = t1 @ w2  (512x768 @ 768x384): 32*24 = 768 tiles
  wmma_gemm_f32<<<96, 256, 0, stream>>>(t1, w2, xm, ROWS, HALF_D, DDIM);

  // 4) xc = [x_m, mean_g x_m]
  build_xc_kernel<<<(ROWS * DDIM) / 256, 256, 0, stream>>>(xm, xc);

  // 5) t2 = xc @ w3   (512x768 @ 768x768)
  wmma_gemm_f32<<<192, 256, 0, stream>>>(xc, w3, t2, ROWS, DDIM, DDIM);

  // 6) s = t2 @ w4
  dot_w4_kernel<<<ROWS, 256, 0, stream>>>(t2, w4, s);

  // 7) top-2 per batch
  top2_kernel<<<1, 64, 0, stream>>>(s, idx);

  // 8) gather selected frames + cls into output
  gather_kernel<<<BDIM * OUTR, 192, 0, stream>>>(inp, idx, out);
}